// TensorBoard_42442866819801
// MI455X (gfx1250) — compile-verified
//
#include <hip/hip_runtime.h>

// Problem constants (match reference)
#define GO_B 256
#define GO_H 19
#define GO_W 19
#define GO_HW 361
#define GO_MAX_MOVES 3610  // HW * 10

// Native clang vector types (required by nontemporal + async-LDS builtins)
typedef int   v4i __attribute__((ext_vector_type(4)));
typedef float v4f __attribute__((ext_vector_type(4)));
typedef __attribute__((address_space(1))) v4i as1_v4i;
typedef __attribute__((address_space(3))) v4i as3_v4i;

// ---- CDNA5 async global->LDS (probe-confirmed present on this toolchain) ----
#if defined(__has_builtin)
#if __has_builtin(__builtin_amdgcn_global_load_async_to_lds_b128) && \
    __has_builtin(__builtin_amdgcn_s_wait_asynccnt)
#define USE_ASYNC_LDS 1
#endif
#endif
#ifndef USE_ASYNC_LDS
#define USE_ASYNC_LDS 0
#endif

// -----------------------------------------------------------------------
// Kernel 1: streaming convert-copy of board_history (int32 -> float32).
// Memory-bound: 1.33 GB in + 1.33 GB out => ~115us floor at 23.3 TB/s.
// b128 per lane; async global->LDS inbound stream (ASYNCcnt), NT stores.
// -----------------------------------------------------------------------
__global__ __launch_bounds__(256) void hist_copy_kernel(
    const int* __restrict__ src, float* __restrict__ dst, long long nvec) {
  const long long i = (long long)blockIdx.x * 256 + threadIdx.x;

  v4i v;
#if USE_ASYNC_LDS
  __shared__ v4i tile[256];
  if (i < nvec) {
    const v4i* sp = (const v4i*)src + i;
    // prefetch ~8KB ahead into L2 (global_prefetch_b8)
    __builtin_prefetch((const void*)(sp + 512), 0, 0);
    // async DMA 16B into this lane's LDS slot (tracked by ASYNCcnt)
    __builtin_amdgcn_global_load_async_to_lds_b128(
        (as1_v4i*)sp, (as3_v4i*)&tile[threadIdx.x], 0, 0);
  }
  __builtin_amdgcn_s_wait_asynccnt(0);
  __asm__ volatile("" ::: "memory");
  if (i >= nvec) return;
  v = tile[threadIdx.x];
#else
  if (i >= nvec) return;
  const v4i* sp = (const v4i*)src + i;
  __builtin_prefetch((const void*)(sp + 512), 0, 0);
  v = __builtin_nontemporal_load(sp);
#endif

  v4f f;
  f.x = (float)v.x;
  f.y = (float)v.y;
  f.z = (float)v.z;
  f.w = (float)v.w;
  __builtin_nontemporal_store(f, (v4f*)dst + i);
}

// -----------------------------------------------------------------------
// Kernel 2: per-batch Go update. One block per batch, 384 threads
// (12 wave32), lanes 0..360 own one board cell each.
// -----------------------------------------------------------------------
__global__ __launch_bounds__(384) void go_update_kernel(
    const float* __restrict__ board, const int* __restrict__ cur_player,
    const int* __restrict__ ko_points, const int* __restrict__ pass_count,
    const int* __restrict__ move_count, const int* __restrict__ positions,
    const int* __restrict__ roots, const int* __restrict__ colour,
    const int* __restrict__ cap_groups, const int* __restrict__ cap_sizes,
    const int* __restrict__ total_caps,
    float* __restrict__ out_board, float* __restrict__ out_hist,
    float* __restrict__ out_ko, float* __restrict__ out_pass,
    float* __restrict__ out_mc, float* __restrict__ out_player) {
  const int b = blockIdx.x;
  const int j = threadIdx.x;

  const int pr = positions[2 * b + 0];
  const int pc = positions[2 * b + 1];
  const bool is_pass = (pr < 0) || (pc < 0);
  const bool play = !is_pass;
  const int cp = cur_player[b];

  const int r = min(max(pr, 0), GO_H - 1);
  const int c = min(max(pc, 0), GO_W - 1);
  const int fp = r * GO_W + c;                    // flat_pos
  const int cell = (b * GO_H + r) * GO_W + c;     // (b,r,c) flat

  // neighbor group roots at the played cell (broadcast via cache)
  const int nr0 = cap_groups[cell * 4 + 0];
  const int nr1 = cap_groups[cell * 4 + 1];
  const int nr2 = cap_groups[cell * 4 + 2];
  const int nr3 = cap_groups[cell * 4 + 3];

  if (j < GO_HW) {
    const float bj = board[(long long)b * GO_HW + j];

    // ---- new_board: place stone, then clear captured groups ----
    float placed = (play && (j == fp)) ? (float)cp : bj;
    const int rt = roots[(long long)b * GO_HW + j];
    const int cl = colour[(long long)b * GO_HW + j];
    const bool hit = (nr0 >= 0 && rt == nr0) || (nr1 >= 0 && rt == nr1) ||
                     (nr2 >= 0 && rt == nr2) || (nr3 >= 0 && rt == nr3);
    const bool cap = play && (cl == (1 - cp)) && hit;
    out_board[(long long)b * GO_HW + j] = cap ? -1.0f : placed;

    // ---- overwrite history row move_count (pre-increment) ----
    const int mv = move_count[b];
    if (mv < GO_MAX_MOVES) {
      const int row = min(max(mv, 0), GO_MAX_MOVES - 1);
      const float bs = (bj == 0.0f) ? 0.0f : ((bj == 1.0f) ? 1.0f : -1.0f);
      out_hist[((long long)b * GO_MAX_MOVES + row) * GO_HW + j] = bs;
    }
  }

  if (j == 0) {
    // ---- ko point ----
    int k0 = is_pass ? ko_points[2 * b + 0] : -1;
    int k1 = is_pass ? ko_points[2 * b + 1] : -1;
    if (play) {
      const bool single_cap = (total_caps[cell] == 1);
      const int s0 = cap_sizes[cell * 4 + 0];
      const int s1 = cap_sizes[cell * 4 + 1];
      const int s2 = cap_sizes[cell * 4 + 2];
      const int s3 = cap_sizes[cell * 4 + 3];
      // argmax of (size==1): first matching index, else 0
      int dir = 0;
      if (s0 == 1) dir = 0;
      else if (s1 == 1) dir = 1;
      else if (s2 == 1) dir = 2;
      else if (s3 == 1) dir = 3;
      const int off = (dir == 0) ? -GO_W : (dir == 1) ? GO_W
                    : (dir == 2) ? -1 : 1;
      const int nbr = fp + off;
      // Python floor division/modulo semantics (nbr may be negative)
      int q = nbr / GO_W;
      int m = nbr % GO_W;
      if (m < 0) { q -= 1; m += GO_W; }
      k0 = single_cap ? q : -1;
      k1 = single_cap ? m : -1;
    }
    out_ko[2 * b + 0] = (float)k0;
    out_ko[2 * b + 1] = (float)k1;

    out_pass[b] = (float)(is_pass ? (pass_count[b] + 1) : 0);
    out_mc[b] = (float)(move_count[b] + 1);
    out_player[b] = (float)(cp ^ 1);
  }
}

// -----------------------------------------------------------------------
extern "C" void kernel_launch(void* const* d_in, const int* in_sizes, int n_in,
                              void* d_out, int out_size, void* d_ws, size_t ws_size,
                              hipStream_t stream) {
  (void)in_sizes; (void)n_in; (void)out_size; (void)d_ws; (void)ws_size;

  const float* board      = (const float*)d_in[0];
  const int*   cur_player = (const int*)d_in[1];
  const int*   ko_points  = (const int*)d_in[2];
  const int*   pass_cnt   = (const int*)d_in[3];
  const int*   hist       = (const int*)d_in[4];
  const int*   move_cnt   = (const int*)d_in[5];
  const int*   positions  = (const int*)d_in[6];
  const int*   roots      = (const int*)d_in[7];
  const int*   colour     = (const int*)d_in[8];
  const int*   cap_groups = (const int*)d_in[9];
  const int*   cap_sizes  = (const int*)d_in[10];
  const int*   total_caps = (const int*)d_in[11];

  // Output layout (tuple return order, all float32):
  float* out        = (float*)d_out;
  float* out_board  = out;                                            // B*HW
  float* out_hist   = out_board + (size_t)GO_B * GO_HW;               // B*MM*HW
  float* out_ko     = out_hist + (size_t)GO_B * GO_MAX_MOVES * GO_HW; // B*2
  float* out_pass   = out_ko + (size_t)2 * GO_B;                      // B
  float* out_mc     = out_pass + GO_B;                                // B
  float* out_player = out_mc + GO_B;                                  // B

  // 1) Bulk history copy (int32 -> float32), 16B vectors.
  const long long total = (long long)GO_B * GO_MAX_MOVES * GO_HW;  // 333,621,760
  const long long nvec = total / 4;                                // divisible
  const int blocks = (int)((nvec + 255) / 256);
  hist_copy_kernel<<<blocks, 256, 0, stream>>>(hist, out_hist, nvec);

  // 2) Per-batch update (also overwrites one history row; same stream orders it).
  go_update_kernel<<<GO_B, 384, 0, stream>>>(
      board, cur_player, ko_points, pass_cnt, move_cnt, positions, roots,
      colour, cap_groups, cap_sizes, total_caps,
      out_board, out_hist, out_ko, out_pass, out_mc, out_player);
}